// PixelContrastLoss_27092653703789
// MI455X (gfx1250) — compile-verified
//
#include <hip/hip_runtime.h>
#include <stdint.h>

// ---- problem constants (from reference) ----
#define TEMP        0.1f
#define BASE_TEMP   0.07f
#define CLASS_NUM   19
#define CACHE_SIZE  2048
#define K_POS       1024u
#define K_NEG       2048u
#define FEAT        256
#define N_VIEW      50
#define A_ROWS      800                       // 16 * 50
#define C_ROWS      (CLASS_NUM * CACHE_SIZE)  // 38912

typedef __attribute__((ext_vector_type(2))) float v2f;
typedef __attribute__((ext_vector_type(8))) float v8f;

// monotonic float <-> uint key mapping (key increases with float value)
__device__ __forceinline__ unsigned f2key(float f) {
  unsigned u = __float_as_uint(f);
  return u ^ ((u >> 31) ? 0xFFFFFFFFu : 0x80000000u);
}
__device__ __forceinline__ float key2f(unsigned k) {
  unsigned u = (k & 0x80000000u) ? (k ^ 0x80000000u) : ~k;
  return __uint_as_float(u);
}

// ---------------------------------------------------------------------------
// Kernel 1: L2-normalize each 256-wide row, write K-PAIRED transpose:
// dstT layout [FEAT/2][rows][2], element (k>>1, r, k&1) = v[k] * scale/|v|.
// This makes each WMMA fragment a single aligned b64 load.
// ---------------------------------------------------------------------------
__global__ void __launch_bounds__(FEAT)
normT_kernel(const float* __restrict__ src, float* __restrict__ dstT,
             int ldT, float scale) {
  __shared__ float red[8];
  const int r = blockIdx.x;
  const int t = threadIdx.x;              // 0..255 == feature index
  float v = src[(size_t)r * FEAT + t];
  float ss = v * v;
  for (int off = 16; off > 0; off >>= 1) ss += __shfl_down(ss, off, 32);
  if ((t & 31) == 0) red[t >> 5] = ss;
  __syncthreads();
  if (t < 8) {
    float s = red[t];
    for (int off = 4; off > 0; off >>= 1) s += __shfl_down(s, off, 8);
    if (t == 0) red[0] = s;
  }
  __syncthreads();
  const float inv = scale / sqrtf(red[0]);
  // [t>>1][r][t&1]
  dstT[(((size_t)(t >> 1) * ldT) + r) * 2 + (t & 1)] = v * inv;
}

// ---------------------------------------------------------------------------
// Kernel 2: adc[800, 38912] = anchor · contrastT via V_WMMA_F32_16X16X4_F32.
// One wave owns a 16x16 output tile; 64 chained K=4 WMMAs (K=256 total).
// A frag (32-bit A 16x4): lanes 0-15 hold K=k0,k0+1 ; lanes 16-31 K=k0+2,k0+3
//   -> pair index k2 = k0/2 + half, one global_load_b64 per fragment.
// ---------------------------------------------------------------------------
__global__ void __launch_bounds__(256)
gemm_wmma_kernel(const float2* __restrict__ aP,  // [FEAT/2][A_ROWS] float2
                 const float2* __restrict__ cP,  // [FEAT/2][C_ROWS] float2
                 float* __restrict__ out) {      // [A_ROWS][C_ROWS]
  const int lane = threadIdx.x & 31;
  const int wave = threadIdx.x >> 5;             // 0..7
  const int half = lane >> 4;                    // 0 or 1
  const int mr   = lane & 15;
  const int row0 = blockIdx.y * 16;              // anchor tile
  const int col0 = (blockIdx.x * 8 + wave) * 16; // contrast tile

  v8f acc = {};
  #pragma unroll 8
  for (int k0 = 0; k0 < FEAT; k0 += 4) {
    const int k2 = (k0 >> 1) + half;
    const float2 av = aP[(size_t)k2 * A_ROWS + row0 + mr];
    const float2 bv = cP[(size_t)k2 * C_ROWS + col0 + mr];
    v2f a, b;
    a.x = av.x; a.y = av.y;
    b.x = bv.x; b.y = bv.y;
    acc = __builtin_amdgcn_wmma_f32_16x16x4_f32(
        /*neg_a=*/false, a, /*neg_b=*/false, b,
        /*c_mod=*/(short)0, acc, /*reuse_a=*/false, /*reuse_b=*/false);
  }
  #pragma unroll
  for (int i = 0; i < 8; ++i) {
    // C/D layout: VGPR i holds M=i (lanes 0-15) and M=i+8 (lanes 16-31)
    out[(size_t)(row0 + i + half * 8) * C_ROWS + col0 + mr] = acc[i];
  }
}

// ---------------------------------------------------------------------------
// Kernel 3: per-anchor exact top-k (by value multiset) + loss.
// Whole 38912-float row lives in LDS (152 KB < 320 KB/WGP), loaded with
// CDNA5 async global->LDS copies. 4-pass radix select over monotonic keys.
// ---------------------------------------------------------------------------
#define SEL_THREADS 512

// Block-cooperative radix select: threshold key T such that exactly
// (K - *ties) elements in the included range are strictly beyond T and
// *ties copies of T complete the selection. largest=true -> K largest.
__device__ unsigned radix_select_block(const float* __restrict__ row,
                                       int incLo, int incHi, int exLo, int exHi,
                                       unsigned K, bool largest,
                                       unsigned* hist, unsigned* ctrl,
                                       unsigned* ties) {
  unsigned prefix = 0, prefmask = 0, krem = K;
  for (int pass = 0; pass < 4; ++pass) {
    const int shift = 24 - 8 * pass;
    for (int i = threadIdx.x; i < 256; i += blockDim.x) hist[i] = 0u;
    __syncthreads();
    for (int j = incLo + (int)threadIdx.x; j < incHi; j += blockDim.x) {
      if (j >= exLo && j < exHi) continue;
      const unsigned k = f2key(row[j]);
      if ((k & prefmask) == prefix)
        atomicAdd(&hist[(k >> shift) & 255u], 1u);
    }
    __syncthreads();
    if (threadIdx.x == 0) {
      unsigned cum = 0; int b;
      if (largest) { for (b = 255; ; --b) { cum += hist[b]; if (cum >= krem) break; } }
      else         { for (b = 0;   ; ++b) { cum += hist[b]; if (cum >= krem) break; } }
      ctrl[0] = (unsigned)b;
      ctrl[1] = krem - (cum - hist[b]);
    }
    __syncthreads();
    prefix  |= ctrl[0] << shift;
    prefmask |= 0xFFu << shift;
    krem = ctrl[1];
    __syncthreads();   // protect ctrl/hist before next pass
  }
  *ties = krem;
  return prefix;
}

__global__ void __launch_bounds__(SEL_THREADS)
select_kernel(const float* __restrict__ adc, const int* __restrict__ yAnchor,
              float* __restrict__ losses) {
  extern __shared__ unsigned char smem[];
  float*    row  = (float*)smem;                                   // C_ROWS f32
  unsigned* hist = (unsigned*)(smem + (size_t)C_ROWS * 4);         // 256 u32
  unsigned* ctrl = hist + 256;                                     // 4 u32
  float*    fctl = (float*)(ctrl + 4);                             // 2 f32

  const int a   = blockIdx.x;
  const int cls = yAnchor[a / N_VIEW];
  const int sl0 = cls * CACHE_SIZE, sl1 = sl0 + CACHE_SIZE;
  const float* g = adc + (size_t)a * C_ROWS;

  // ---- async copy: global row -> LDS (B128 per lane) ----
  const unsigned ldsBase = (unsigned)(size_t)row;
  for (int i = threadIdx.x; i < C_ROWS / 4; i += SEL_THREADS) {
    const unsigned lds = ldsBase + (unsigned)i * 16u;
    const unsigned long long ga = (unsigned long long)(size_t)(g + i * 4);
    asm volatile("global_load_async_to_lds_b128 %0, %1, off"
                 :: "v"(lds), "v"(ga) : "memory");
  }
  asm volatile("s_wait_asynccnt 0" ::: "memory");
  __syncthreads();

  // ---- thresholds: 2048 largest outside slab, 1024 smallest inside slab ----
  unsigned tieN, tieP;
  const unsigned TnKey =
      radix_select_block(row, 0, C_ROWS, sl0, sl1, K_NEG, true,  hist, ctrl, &tieN);
  const unsigned TpKey =
      radix_select_block(row, sl0, sl1, -1, -1,    K_POS, false, hist, ctrl, &tieP);

  if (threadIdx.x == 0) { ctrl[2] = 0u; fctl[0] = 0.f; fctl[1] = 0.f; }
  __syncthreads();

  // ---- max over selected negatives == global max outside slab ----
  unsigned mk = 0;
  for (int j = threadIdx.x; j < C_ROWS; j += blockDim.x) {
    if (j >= sl0 && j < sl1) continue;
    const unsigned k = f2key(row[j]);
    if (k > mk) mk = k;
  }
  atomicMax(&ctrl[2], mk);
  __syncthreads();
  // max selected pos == pos threshold value (>=1 copy always selected)
  const float m = fmaxf(key2f(TpKey), key2f(ctrl[2]));

  // ---- S_neg = sum exp over selected negatives ----
  float s = 0.f;
  for (int j = threadIdx.x; j < C_ROWS; j += blockDim.x) {
    if (j >= sl0 && j < sl1) continue;
    const float v = row[j];
    if (f2key(v) > TnKey) s += expf(v - m);
  }
  atomicAdd(&fctl[0], s);
  __syncthreads();
  const float Sneg = fctl[0] + (float)tieN * expf(key2f(TnKey) - m);

  // ---- pos contributions: (v-m) - log(exp(v-m) + S_neg) ----
  float p = 0.f;
  for (int j = sl0 + (int)threadIdx.x; j < sl1; j += blockDim.x) {
    const float v = row[j];
    if (f2key(v) < TpKey) {
      const float l = v - m;
      p += l - logf(expf(l) + Sneg);
    }
  }
  atomicAdd(&fctl[1], p);
  __syncthreads();
  if (threadIdx.x == 0) {
    const float lt  = key2f(TpKey) - m;
    const float tot = fctl[1] + (float)tieP * (lt - logf(expf(lt) + Sneg));
    losses[a] = -(TEMP / BASE_TEMP) * tot / (float)K_POS;
  }
}

// ---------------------------------------------------------------------------
// Kernel 4: mean over 800 anchor losses -> d_out[0]
// ---------------------------------------------------------------------------
__global__ void __launch_bounds__(256)
reduce_kernel(const float* __restrict__ losses, float* __restrict__ out) {
  __shared__ float red[8];
  float s = 0.f;
  for (int i = threadIdx.x; i < A_ROWS; i += 256) s += losses[i];
  for (int off = 16; off > 0; off >>= 1) s += __shfl_down(s, off, 32);
  if ((threadIdx.x & 31) == 0) red[threadIdx.x >> 5] = s;
  __syncthreads();
  if (threadIdx.x == 0) {
    float t = 0.f;
    #pragma unroll
    for (int i = 0; i < 8; ++i) t += red[i];
    out[0] = t / (float)A_ROWS;
  }
}

// ---------------------------------------------------------------------------
extern "C" void kernel_launch(void* const* d_in, const int* in_sizes, int n_in,
                              void* d_out, int out_size, void* d_ws, size_t ws_size,
                              hipStream_t stream) {
  const float* X = (const float*)d_in[0];   // [16,50,256]
  const float* Q = (const float*)d_in[1];   // [19,2048,256]
  const int*   y = (const int*)d_in[2];     // [16]
  float* outp = (float*)d_out;

  char* ws = (char*)d_ws;
  size_t off = 0;
  auto alloc = [&](size_t bytes) -> void* {
    void* p = ws + off;
    off = (off + bytes + 255) & ~(size_t)255;
    return p;
  };
  float* anchorT   = (float*)alloc((size_t)FEAT * A_ROWS * 4);  // [128][800] float2
  float* contrastT = (float*)alloc((size_t)FEAT * C_ROWS * 4);  // [128][38912] float2
  float* adc       = (float*)alloc((size_t)A_ROWS * C_ROWS * 4);// [800][38912]
  float* losses    = (float*)alloc((size_t)A_ROWS * 4);

  // 1) normalize (+1/TEMP on contrast side) into K-paired transposed layout
  normT_kernel<<<A_ROWS, FEAT, 0, stream>>>(X, anchorT, A_ROWS, 1.0f);
  normT_kernel<<<C_ROWS, FEAT, 0, stream>>>(Q, contrastT, C_ROWS, 1.0f / TEMP);

  // 2) fp32 WMMA GEMM -> logits matrix (L2-resident, 124 MB)
  gemm_wmma_kernel<<<dim3(C_ROWS / 128, A_ROWS / 16), 256, 0, stream>>>(
      (const float2*)anchorT, (const float2*)contrastT, adc);

  // 3) per-anchor selection + loss (row in LDS via async copy)
  const size_t shmem = (size_t)C_ROWS * 4 + 256 * 4 + 64;
  select_kernel<<<A_ROWS, SEL_THREADS, shmem, stream>>>(adc, y, losses);

  // 4) mean
  reduce_kernel<<<1, 256, 0, stream>>>(losses, outp);
}